// VPmatrixPointsV1_15187004359121
// MI455X (gfx1250) — compile-verified
//
#include <hip/hip_runtime.h>

#define HH 512
#define WW 512
#define HWP (HH * WW)

typedef __attribute__((ext_vector_type(2))) float v2f;
typedef __attribute__((ext_vector_type(8))) float v8f;

// ---------------------------------------------------------------------------
// Stage 0: VP[b] = P[b] @ V[b]   (64 batches of 4x4)
// ---------------------------------------------------------------------------
__global__ void vp_kernel(const float* __restrict__ V, const float* __restrict__ P,
                          float* __restrict__ VP, int nb) {
    int t = blockIdx.x * blockDim.x + threadIdx.x;
    if (t >= nb * 16) return;
    int b = t >> 4, ij = t & 15, i = ij >> 2, j = ij & 3;
    const float* Pb = P + b * 16;
    const float* Vb = V + b * 16;
    float s = 0.f;
#pragma unroll
    for (int k = 0; k < 4; ++k) s += Pb[i * 4 + k] * Vb[k * 4 + j];
    VP[t] = s;
}

// ---------------------------------------------------------------------------
// Stage 1: zero plane0 (the splat target) of every batch, float4 stores
// ---------------------------------------------------------------------------
__global__ void zero_plane0(float* __restrict__ out, int nb) {
    size_t q = (size_t)blockIdx.x * blockDim.x + threadIdx.x;
    size_t nq = (size_t)nb * (HWP / 4);
    if (q >= nq) return;
    size_t b = q / (HWP / 4), r = q % (HWP / 4);
    ((float4*)(out + b * (size_t)3 * HWP))[r] = make_float4(0.f, 0.f, 0.f, 0.f);
}

// ---------------------------------------------------------------------------
// Stage 2: WMMA point transform + splat.
// One wave = 16 points x 4 batches via V_WMMA_F32_16X16X4_F32:
//   A (16x4)  = points (M=point, K=coord)
//   B (4x16)  = 4 VP matrices packed column-wise: B[i][4*bb+j] = VP[bb][j][i]
//   D (16x16) : D[p][4*bb+j] = tp[bb][p][j]
// D is staged through LDS so each lane can grab x,y,w of one (point,batch).
// ---------------------------------------------------------------------------
__global__ void __launch_bounds__(32)
splat_kernel(const float* __restrict__ pts, const float* __restrict__ VP,
             float* __restrict__ out, int npts) {
    __shared__ float VPsh[64];   // 4 matrices: VPsh[bb*16 + j*4 + i]
    __shared__ float Dsh[256];   // D[M][N] row-major

    const int lane = threadIdx.x;      // 0..31
    const int tile = blockIdx.x;       // point tile (16 points)
    const int bg   = blockIdx.y;       // batch group (4 batches)

    VPsh[lane]      = VP[bg * 64 + lane];
    VPsh[lane + 32] = VP[bg * 64 + lane + 32];
    __syncthreads();

    const int half = lane >> 4;        // 0: K={0,1}, 1: K={2,3}
    const int col  = lane & 15;
    const int k0   = half * 2;

    // A fragment: point M = col
    int p = tile * 16 + col;
    int pc = (p < npts) ? p : (npts - 1);
    const float* p4 = pts + (size_t)pc * 4;
    v2f a;
    a.x = p4[k0];
    a.y = p4[k0 + 1];

    // B fragment: column N = col encodes (bb = col>>2, j = col&3); rows K = k0, k0+1
    const int bb = col >> 2, jj = col & 3;
    v2f bf;
    bf.x = VPsh[bb * 16 + jj * 4 + k0];
    bf.y = VPsh[bb * 16 + jj * 4 + k0 + 1];

    v8f c = {};
    c = __builtin_amdgcn_wmma_f32_16x16x4_f32(false, a, false, bf,
                                              (short)0, c, false, false);

#pragma unroll
    for (int v = 0; v < 8; ++v) Dsh[(v + half * 8) * 16 + col] = c[v];
    __syncthreads();

    // 64 (point,batch) pairs handled by 32 lanes in 2 rounds
#pragma unroll
    for (int it = 0; it < 2; ++it) {
        int idx = it * 32 + lane;          // 0..63
        int pl = idx & 15;                 // local point
        int bl = idx >> 4;                 // local batch
        int gp = tile * 16 + pl;
        if (gp >= npts) continue;
        float x = Dsh[pl * 16 + bl * 4 + 0];
        float y = Dsh[pl * 16 + bl * 4 + 1];
        float w = Dsh[pl * 16 + bl * 4 + 3];
        float xd = (w != 0.f) ? x / w : x;
        float yd = (w != 0.f) ? y / w : y;
        int sx = (int)rintf((xd + 1.f) * 0.5f * (float)WW);
        int sy = (int)rintf((1.f - (yd + 1.f) * 0.5f) * (float)HH);
        if (sx >= 0 && sx < WW && sy >= 0 && sy < HH) {
            int gb = bg * 4 + bl;
            out[(size_t)gb * 3 * HWP + (size_t)sy * WW + sx] = 255.0f;
        }
    }
}

// ---------------------------------------------------------------------------
// Separable 9x9 max dilation (OOB = -inf  ==> just skip out-of-range taps)
// src/dst pointers carry the channel offset; per-batch stride is 3*H*W.
// ---------------------------------------------------------------------------
__global__ void dilate_h(const float* __restrict__ src, float* __restrict__ dst, int nb) {
    size_t i = (size_t)blockIdx.x * blockDim.x + threadIdx.x;
    if (i >= (size_t)nb * HWP) return;
    int b = (int)(i / HWP), pix = (int)(i % HWP);
    int y = pix / WW, x = pix % WW;
    const float* row = src + (size_t)b * 3 * HWP + (size_t)y * WW;
    int x0 = x - 4; if (x0 < 0) x0 = 0;
    int x1 = x + 4; if (x1 > WW - 1) x1 = WW - 1;
    float m = -__builtin_inff();
    for (int xx = x0; xx <= x1; ++xx) m = fmaxf(m, row[xx]);
    dst[(size_t)b * 3 * HWP + (size_t)y * WW + x] = m;
}

__global__ void dilate_v(const float* __restrict__ src, float* __restrict__ dst, int nb) {
    size_t i = (size_t)blockIdx.x * blockDim.x + threadIdx.x;
    if (i >= (size_t)nb * HWP) return;
    int b = (int)(i / HWP), pix = (int)(i % HWP);
    int y = pix / WW, x = pix % WW;
    const float* img = src + (size_t)b * 3 * HWP;
    int y0 = y - 4; if (y0 < 0) y0 = 0;
    int y1 = y + 4; if (y1 > HH - 1) y1 = HH - 1;
    float m = -__builtin_inff();
    for (int yy = y0; yy <= y1; ++yy) m = fmaxf(m, img[(size_t)yy * WW + x]);
    dst[(size_t)b * 3 * HWP + (size_t)y * WW + x] = m;
}

// ---------------------------------------------------------------------------
// Separable 9-tap Gaussian, reflect padding (np.pad 'reflect': edge excluded)
// sigma = 0.3*((9-1)*0.5-1)+0.8 = 1.7, normalized taps precomputed:
// ---------------------------------------------------------------------------
__device__ __constant__ float GK[9] = {
    0.0148394f, 0.0498173f, 0.1183222f, 0.1988296f, 0.2363830f,
    0.1988296f, 0.1183222f, 0.0498173f, 0.0148394f};

__global__ void blur_h(const float* __restrict__ src, float* __restrict__ dst, int nb) {
    size_t i = (size_t)blockIdx.x * blockDim.x + threadIdx.x;
    if (i >= (size_t)nb * HWP) return;
    int b = (int)(i / HWP), pix = (int)(i % HWP);
    int y = pix / WW, x = pix % WW;
    const float* row = src + (size_t)b * 3 * HWP + (size_t)y * WW;
    float s = 0.f;
#pragma unroll
    for (int k = 0; k < 9; ++k) {
        int xx = x + k - 4;
        if (xx < 0) xx = -xx;
        if (xx > WW - 1) xx = 2 * (WW - 1) - xx;
        s += GK[k] * row[xx];
    }
    dst[(size_t)b * 3 * HWP + (size_t)y * WW + x] = s;
}

__global__ void blur_v(const float* __restrict__ src, float* __restrict__ dst, int nb) {
    size_t i = (size_t)blockIdx.x * blockDim.x + threadIdx.x;
    if (i >= (size_t)nb * HWP) return;
    int b = (int)(i / HWP), pix = (int)(i % HWP);
    int y = pix / WW, x = pix % WW;
    const float* img = src + (size_t)b * 3 * HWP;
    float s = 0.f;
#pragma unroll
    for (int k = 0; k < 9; ++k) {
        int yy = y + k - 4;
        if (yy < 0) yy = -yy;
        if (yy > HH - 1) yy = 2 * (HH - 1) - yy;
        s += GK[k] * img[(size_t)yy * WW + x];
    }
    dst[(size_t)b * 3 * HWP + (size_t)y * WW + x] = s;
}

// ---------------------------------------------------------------------------
// Threshold + broadcast to 3 channels, float4 stores (the only HBM-heavy pass)
// Reads plane2, writes planes 0,1,2 at the SAME pixel -> no cross-thread hazard.
// ---------------------------------------------------------------------------
__global__ void thresh_bcast(float* __restrict__ out, int nb) {
    size_t q = (size_t)blockIdx.x * blockDim.x + threadIdx.x;
    size_t nq = (size_t)nb * (HWP / 4);
    if (q >= nq) return;
    size_t b = q / (HWP / 4), r = q % (HWP / 4);
    float* base = out + b * (size_t)3 * HWP;
    float4 v = ((const float4*)(base + 2 * (size_t)HWP))[r];
    float4 m;
    m.x = (rintf(v.x) > 100.f) ? 1.f : 0.f;
    m.y = (rintf(v.y) > 100.f) ? 1.f : 0.f;
    m.z = (rintf(v.z) > 100.f) ? 1.f : 0.f;
    m.w = (rintf(v.w) > 100.f) ? 1.f : 0.f;
    ((float4*)(base + 0 * (size_t)HWP))[r] = m;
    ((float4*)(base + 1 * (size_t)HWP))[r] = m;
    ((float4*)(base + 2 * (size_t)HWP))[r] = m;
}

// ---------------------------------------------------------------------------
extern "C" void kernel_launch(void* const* d_in, const int* in_sizes, int n_in,
                              void* d_out, int out_size, void* d_ws, size_t ws_size,
                              hipStream_t stream) {
    const float* V   = (const float*)d_in[0];
    const float* P   = (const float*)d_in[1];
    const float* pts = (const float*)d_in[2];
    float* out = (float*)d_out;
    float* VPbuf = (float*)d_ws;                 // 64*16 floats = 4 KB scratch

    const int nb   = in_sizes[0] / 16;           // 64 batches
    const int npts = in_sizes[2] / 4;            // 13860 points

    // VP = P @ V
    vp_kernel<<<(nb * 16 + 255) / 256, 256, 0, stream>>>(V, P, VPbuf, nb);

    // zero splat plane (channel 0 of every batch)
    {
        size_t nq = (size_t)nb * (HWP / 4);
        zero_plane0<<<(unsigned)((nq + 255) / 256), 256, 0, stream>>>(out, nb);
    }

    // WMMA transform + splat into plane0
    {
        dim3 grid((npts + 15) / 16, nb / 4);
        splat_kernel<<<grid, 32, 0, stream>>>(pts, VPbuf, out, npts);
    }

    size_t npix = (size_t)nb * HWP;
    unsigned gpix = (unsigned)((npix + 255) / 256);

    // plane0 -> plane1 -> plane2 -> plane1 -> plane2 (L2-resident passes)
    dilate_h<<<gpix, 256, 0, stream>>>(out + 0 * HWP, out + 1 * HWP, nb);
    dilate_v<<<gpix, 256, 0, stream>>>(out + 1 * HWP, out + 2 * HWP, nb);
    blur_h  <<<gpix, 256, 0, stream>>>(out + 2 * HWP, out + 1 * HWP, nb);
    blur_v  <<<gpix, 256, 0, stream>>>(out + 1 * HWP, out + 2 * HWP, nb);

    // threshold + broadcast (final 201 MB HBM write)
    {
        size_t nq = (size_t)nb * (HWP / 4);
        thresh_bcast<<<(unsigned)((nq + 255) / 256), 256, 0, stream>>>(out, nb);
    }
}